// Stack_79061757984791
// MI455X (gfx1250) — compile-verified
//
#include <hip/hip_runtime.h>
#include <cstdint>

// ---------------------------------------------------------------------------
// TMM (transfer-matrix method) for s/p polarization on gfx1250.
//   out[0] = R, out[1] = T, out[2] = 1-(R+T), each (A, W) fp32.
// Compute-bound (fp32 VALU + transcendentals); memory floor ~0.55us @23.3TB/s.
// CDNA5 paths: TDM tensor_load_to_lds (TENSORcnt) stages the refractive-index
// slab into LDS; s_wait_tensorcnt; wave32 block shaping; NT streaming stores.
// WMMA intentionally NOT used: no GEMM structure (sequential 2x2 complex
// chains per point, fp32 precision mandated by the complex64 reference).
// ---------------------------------------------------------------------------

#define TS      256   // wavelengths per block == blockDim.x (8 wave32 waves)
#define MAX_NL  18    // layers+2 in the reference setup (L=16)
#define TWO_PI_F 6.283185307179586f

struct Cf { float re, im; };
__device__ __forceinline__ Cf cmul(Cf a, Cf b) {
  return { a.re * b.re - a.im * b.im, a.re * b.im + a.im * b.re };
}
__device__ __forceinline__ Cf cadd(Cf a, Cf b) { return { a.re + b.re, a.im + b.im }; }
__device__ __forceinline__ Cf csub(Cf a, Cf b) { return { a.re - b.re, a.im - b.im }; }
__device__ __forceinline__ Cf cinv(Cf a) {
  float d = __builtin_amdgcn_rcpf(a.re * a.re + a.im * a.im);
  return { a.re * d, -a.im * d };
}

// ---------------------------------------------------------------------------
// Tensor Data Mover: 2D tile (tile_h x tile_w fp32, row stride = row_stride
// elements) from global into LDS. Descriptor packing per CDNA5 ISA section 8.
// ---------------------------------------------------------------------------
#if defined(__HIP_DEVICE_COMPILE__) && __has_builtin(__builtin_amdgcn_tensor_load_to_lds)
#define HAVE_TDM 1
typedef unsigned int tdm_u32x4 __attribute__((ext_vector_type(4)));
typedef int          tdm_i32x4 __attribute__((ext_vector_type(4)));
typedef int          tdm_i32x8 __attribute__((ext_vector_type(8)));

__device__ __forceinline__ void tdm_load_2d_to_lds(
    uint32_t lds_off, uint64_t gaddr,
    uint32_t tensor_w, uint32_t tensor_h,
    uint32_t tile_w, uint32_t tile_h, uint32_t row_stride)
{
  // Group 0: count=1 | lds_addr | global_addr[56:0] | type=2 (bits 127:126)
  tdm_u32x4 g0 = { 1u,
                   lds_off,
                   (uint32_t)gaddr,
                   (uint32_t)((gaddr >> 32) & 0x01FFFFFFu) | (2u << 30) };
  // Group 1: data_size=2 (4B) @bits17:16; tensor_dim0 @79:48; tensor_dim1
  // @111:80; tile_dim0 @127:112; tile_dim1 @143:128; dim0_stride @207:160.
  tdm_i32x8 g1 = { (int)0x00020000u,
                   (int)((tensor_w & 0xFFFFu) << 16),
                   (int)(((tensor_w >> 16) & 0xFFFFu) | ((tensor_h & 0xFFFFu) << 16)),
                   (int)(((tensor_h >> 16) & 0xFFFFu) | ((tile_w & 0xFFFFu) << 16)),
                   (int)(tile_h & 0xFFFFu),
                   (int)row_stride,
                   0, 0 };
  tdm_i32x4 g2 = { 0, 0, 0, 0 };   // 2D tensor: dims 2..4 unused
  tdm_i32x4 g3 = { 0, 0, 0, 0 };
#if defined(__clang_major__) && (__clang_major__ >= 23)
  tdm_i32x8 g4 = { 0, 0, 0, 0, 0, 0, 0, 0 };
  __builtin_amdgcn_tensor_load_to_lds(g0, g1, g2, g3, g4, 0);
#else
  __builtin_amdgcn_tensor_load_to_lds(g0, g1, g2, g3, 0);
#endif
}
#else
#define HAVE_TDM 0
#endif

// ---------------------------------------------------------------------------
// Per-point TMM for one polarization. sn = LDS slab, sn[j*TS + tid] = n_j.
// POL==0 -> 's', POL==1 -> 'p'. NLC > 0: compile-time layer count (full
// unroll of the serial chain so trans-ops interleave); NLC == 0: runtime.
// ---------------------------------------------------------------------------
template <int POL, int NLC>
__device__ __forceinline__ void tmm_eval(
    const float* __restrict__ sn, int tid, const float* __restrict__ thick,
    float lam, float sin0, int nl_rt, float& Rout, float& Tout)
{
  const int NL = (NLC > 0) ? NLC : nl_rt;

  const float n0    = sn[tid];
  const float n0sin = n0 * sin0;
  const float k0    = TWO_PI_F * __builtin_amdgcn_rcpf(lam);

  float nA, crA, ciA, nB, crB, ciB;

  // cosq = sqrt(1 - (n0 sin0 / n)^2); n real -> cosq purely real or imaginary
  // (principal branch: sqrt(negative + 0j) = +i*sqrt(|.|)).
  auto cosq = [&](float n, float& cr, float& ci) {
    float s  = n0sin * __builtin_amdgcn_rcpf(n);
    float c2 = 1.0f - s * s;
    float rt = __builtin_amdgcn_sqrtf(fabsf(c2));
    bool pos = (c2 >= 0.0f);
    cr = pos ? rt : 0.0f;
    ci = pos ? 0.0f : rt;
  };

  nA = n0; cosq(nA, crA, ciA);
  const float q0re = nA * crA;                 // Re(n0 cos0) for T norm
  nB = sn[TS + tid]; cosq(nB, crB, ciB);

  // Fresnel r, t at interface between layer A and layer B.
  auto iface = [&](Cf& r, Cf& t) {
    Cf num, den, tn;
    if (POL == 0) {                            // s-pol
      Cf qa = { nA * crA, nA * ciA };
      Cf qb = { nB * crB, nB * ciB };
      den = cadd(qa, qb); num = csub(qa, qb);
      tn  = { 2.0f * qa.re, 2.0f * qa.im };
    } else {                                   // p-pol
      Cf x = { nB * crA, nB * ciA };           // n_{j+1} cos_j
      Cf y = { nA * crB, nA * ciB };           // n_j cos_{j+1}
      den = cadd(x, y); num = csub(x, y);
      tn  = { 2.0f * nA * crA, 2.0f * nA * ciA };
    }
    Cf invd = cinv(den);
    r = cmul(num, invd);
    t = cmul(tn,  invd);
  };

  Cf r, t;
  iface(r, t);
  Cf m00 = cinv(t);                            // M = (1/t0) [[1, r0],[r0, 1]]
  Cf m11 = m00;
  Cf m01 = cmul(r, m00);
  Cf m10 = m01;

#pragma unroll
  for (int j = 1; j <= NL - 2; ++j) {          // inner layers 1..L
    nA = nB; crA = crB; ciA = ciB;
    nB = sn[(j + 1) * TS + tid]; cosq(nB, crB, ciB);
    iface(r, t);

    // delta = k0 * d_{j-1} * n_j cos_j ; em = exp(-i delta), ep = exp(+i delta)
    const float kd = k0 * thick[j - 1];
    const float dr = kd * nA * crA;
    const float di = kd * nA * ciA;
    float ss, cc;
    __sincosf(dr, &ss, &cc);
    const float eg = __expf(di);
    const float es = __builtin_amdgcn_rcpf(eg);
    const Cf em = { eg * cc, -eg * ss };
    const Cf ep = { es * cc,  es * ss };

    const Cf aa  = cmul(m00, em), bb = cmul(m01, ep);
    const Cf ccm = cmul(m10, em), ee = cmul(m11, ep);
    const Cf it  = cinv(t);
    m00 = cmul(cadd(aa,  cmul(bb, r)), it);
    m01 = cmul(cadd(cmul(aa, r), bb), it);
    m10 = cmul(cadd(ccm, cmul(ee, r)), it);
    m11 = cmul(cadd(cmul(ccm, r), ee), it);
  }

  const float inv00 = __builtin_amdgcn_rcpf(m00.re * m00.re + m00.im * m00.im);
  Rout = (m10.re * m10.re + m10.im * m10.im) * inv00;
  // norm = Re(n_last cos_last)/Re(n0 cos0) (identical for s and p since n real)
  Tout = inv00 * (nB * crB) * __builtin_amdgcn_rcpf(q0re);
}

// ---------------------------------------------------------------------------
// Main kernel: one block per (angle, TS-wavelength tile).
// ---------------------------------------------------------------------------
template <int NLC>
__global__ __launch_bounds__(TS)
void tmm_main_kernel(const float* __restrict__ ntab,
                     const float* __restrict__ data,
                     const int* __restrict__ mat,
                     const float* __restrict__ thick,
                     const float* __restrict__ wl,
                     const float* __restrict__ ang,
                     const int* __restrict__ polp,
                     float* __restrict__ out,
                     int W, int A, int NL, int use_tdm)
{
  __shared__ float sn[MAX_NL * TS];
  const int tid = threadIdx.x;
  const int wt  = blockIdx.x;
  const int a   = blockIdx.y;
  const int w   = wt * TS + tid;
  const int wc  = (w < W) ? w : (W - 1);
  const int nl  = (NLC > 0) ? NLC : ((NL <= MAX_NL) ? NL : MAX_NL);

#if HAVE_TDM
  if (use_tdm) {
    if (tid < warpSize) {          // single wave issues the block's DMA
      const uint32_t lds_off = (uint32_t)(uintptr_t)(&sn[0]); // low32 of flat = LDS offset
      const uint64_t gaddr   = (uint64_t)(uintptr_t)(ntab + (size_t)wt * TS);
      tdm_load_2d_to_lds(lds_off, gaddr,
                         (uint32_t)W, (uint32_t)nl,   // tensor dims (elems)
                         (uint32_t)TS, (uint32_t)nl,  // tile dims
                         (uint32_t)W);                // row stride (elems)
#if __has_builtin(__builtin_amdgcn_s_wait_tensorcnt)
      __builtin_amdgcn_s_wait_tensorcnt(0);
#else
      asm volatile("s_wait_tensorcnt 0x0" ::: "memory");
#endif
    }
  } else
#endif
  {
    (void)use_tdm;
    for (int j = 0; j < nl; ++j)
      sn[j * TS + tid] = (use_tdm ? ntab[(size_t)j * W + wc]
                                  : data[(size_t)mat[j] * W + wc]);
  }
  __syncthreads();

  const float lam  = wl[wc];
  const float sin0 = __sinf(ang[a]);
  const int   pol  = *polp;        // uniform scalar load; wave-uniform branch

  float R, T;
  if (pol == 0) {
    tmm_eval<0, NLC>(sn, tid, thick, lam, sin0, nl, R, T);
  } else if (pol == 1) {
    tmm_eval<1, NLC>(sn, tid, thick, lam, sin0, nl, R, T);
  } else {
    float Rs, Ts, Rp, Tp;
    tmm_eval<0, NLC>(sn, tid, thick, lam, sin0, nl, Rs, Ts);
    tmm_eval<1, NLC>(sn, tid, thick, lam, sin0, nl, Rp, Tp);
    R = 0.5f * (Rs + Rp);
    T = 0.5f * (Ts + Tp);
  }

  if (w < W) {
    const size_t AW = (size_t)A * (size_t)W;
    const size_t o  = (size_t)a * (size_t)W + (size_t)w;
    // Pure streaming output (written once, never re-read): non-temporal.
    __builtin_nontemporal_store(R,                 out + o);
    __builtin_nontemporal_store(T,                 out + AW + o);
    __builtin_nontemporal_store(1.0f - (R + T),    out + 2 * AW + o);
  }
}

// ---------------------------------------------------------------------------
// Gather kernel: n = data[material_distribution] into workspace (NL x W).
// ---------------------------------------------------------------------------
__global__ void tmm_gather_n_kernel(const float* __restrict__ data,
                                    const int* __restrict__ mat,
                                    float* __restrict__ ntab, int W)
{
  const int j = blockIdx.y;
  const int w = blockIdx.x * blockDim.x + threadIdx.x;
  if (w < W) ntab[(size_t)j * W + w] = data[(size_t)mat[j] * W + w];
}

// ---------------------------------------------------------------------------
// Inputs: 0 data(M,W) f32 | 1 thickness(L) f32 | 2 wavelengths(W) f32 |
//         3 angles(A) f32 | 4 material_distribution(L+2) i32 |
//         5 coherency(L) i32 (unused by reference) | 6 polarization scalar i32
// Output: (3, A, W) f32
// ---------------------------------------------------------------------------
extern "C" void kernel_launch(void* const* d_in, const int* in_sizes, int n_in,
                              void* d_out, int out_size, void* d_ws, size_t ws_size,
                              hipStream_t stream)
{
  const float* data  = (const float*)d_in[0];
  const float* thick = (const float*)d_in[1];
  const float* wl    = (const float*)d_in[2];
  const float* ang   = (const float*)d_in[3];
  const int*   mat   = (const int*)d_in[4];
  const int*   polp  = (const int*)d_in[6];
  const int W  = in_sizes[2];
  const int A  = in_sizes[3];
  const int NL = in_sizes[4];          // L + 2
  float* out  = (float*)d_out;
  float* ntab = (float*)d_ws;
  (void)n_in; (void)out_size;

  const size_t ntab_bytes = (size_t)NL * (size_t)W * sizeof(float);
  const int use_tdm = (ws_size >= ntab_bytes && NL <= MAX_NL) ? 1 : 0;

  if (use_tdm) {
    dim3 gg((unsigned)((W + 255) / 256), (unsigned)NL);
    tmm_gather_n_kernel<<<gg, 256, 0, stream>>>(data, mat, ntab, W);
  }
  dim3 grid((unsigned)((W + TS - 1) / TS), (unsigned)A);
  if (NL == MAX_NL) {
    // Reference configuration (L=16): fully-unrolled serial chain.
    tmm_main_kernel<MAX_NL><<<grid, TS, 0, stream>>>(
        ntab, data, mat, thick, wl, ang, polp, out, W, A, NL, use_tdm);
  } else {
    tmm_main_kernel<0><<<grid, TS, 0, stream>>>(
        ntab, data, mat, thick, wl, ang, polp, out, W, A, NL, use_tdm);
  }
}